// SchNetModel_45320494907957
// MI455X (gfx1250) — compile-verified
//
#include <hip/hip_runtime.h>
#include <hip/hip_bf16.h>

typedef __attribute__((ext_vector_type(2))) float v2f;
typedef __attribute__((ext_vector_type(8))) float v8f;

#define NA       24        // atoms per molecule
#define HID      128
#define NS       576       // padded edge slots = 24*24
#define NST      36        // 16-row slot tiles
#define STRIDE   132       // padded row stride for activation tiles in LDS
#define USTR     36        // padded stride for U staging
#define NTHREADS 512
#define NWAVES   16

// LDS layout (float offsets)
#define SH_OFF   0
#define SXV_OFF  (32*STRIDE)               // 4224
#define SAGG_OFF (SXV_OFF + 32*STRIDE)     // 8448
#define SD_OFF   (SAGG_OFF + 32*STRIDE)    // 12672
#define SC_OFF   (SD_OFF + NS)             // 13248
#define SP_OFF   (SC_OFF + NS)             // 13824
#define SU_OFF   (SP_OFF + 128)            // 13952
#define WB0_OFF  (SU_OFF + NWAVES*16*USTR) // 23168 : weight buffer 0 (45 k-tiles)
#define WB1_OFF  (WB0_OFF + 45*512)        // 46208 : weight buffer 1 (45 k-tiles)
#define SMEM_TOT (WB1_OFF + 45*512)        // 69248 floats = 276992 B (~270.5 KB)

__device__ __forceinline__ v8f wmma4(v2f a, v2f b, v8f c) {
    // D = A(16x4,f32) * B(4x16,f32) + C(16x16,f32)
    return __builtin_amdgcn_wmma_f32_16x16x4_f32(false, a, false, b, (short)0, c,
                                                 false, false);
}

__device__ __forceinline__ float sspf(float x) {
    // shifted softplus: log1p(exp(x)) - log(2), numerically stable
    float e = __expf(-fabsf(x));
    return fmaxf(x, 0.0f) + log1pf(e) - 0.69314718056f;
}

__device__ __forceinline__ float rbff(float d, int g) {
    // gaussian rbf; g may exceed 49 (padded K) - those hit zero B rows anyway
    float t = d - (float)g * 0.2040816327f;          // offset step = 10/49
    return __expf(-12.005f * t * t);                 // coeff = -0.5/(10/49)^2
}

// fragment-packed index for weight element (k, n)
__device__ __forceinline__ int fragidx(int k, int n) {
    return ((((k >> 2) * 8) + (n >> 4)) << 6)            // (ktile*8 + ntile)*64
         + ((((n & 15)) + (((k >> 1) & 1) << 4)) << 1)   // lane*2
         + (k & 1);                                      // .x / .y
}

// Stage a full [ktiles*4, 128] matrix (no bounds check), float4 global reads.
__device__ __forceinline__ void stage_w_full(float* dst, const float* __restrict__ src,
                                             int ktiles, int tid) {
    int total = ktiles * 128;                 // in float4 units
    for (int idx = tid; idx < total; idx += NTHREADS) {
        int k  = idx >> 5;                    // 32 float4 per row of 128
        int n4 = (idx & 31) << 2;
        float4 v = *(const float4*)(src + k * HID + n4);
        int base = fragidx(k, n4);            // n4..n4+3 stay in same 16-block
        dst[base]     = v.x;
        dst[base + 2] = v.y;
        dst[base + 4] = v.z;
        dst[base + 6] = v.w;
    }
}

// Stage w1 [50,128] into 13 k-tiles (52 rows); branch-free: pad rows zero-filled.
__device__ __forceinline__ void stage_w1(float* dst, const float* __restrict__ src,
                                         int tid) {
    for (int idx = tid; idx < 50 * 32; idx += NTHREADS) {   // float4 units
        int k  = idx >> 5;
        int n4 = (idx & 31) << 2;
        float4 v = *(const float4*)(src + k * HID + n4);
        int base = fragidx(k, n4);
        dst[base]     = v.x;
        dst[base + 2] = v.y;
        dst[base + 4] = v.z;
        dst[base + 6] = v.w;
    }
    for (int idx = tid; idx < 2 * HID; idx += NTHREADS) {   // rows 50,51 = 0
        int k = 50 + (idx >> 7);
        int n = idx & 127;
        dst[fragidx(k, n)] = 0.0f;
    }
}

// 16x16 output tile of [32,128](LDS, stride 132) @ [128,128](fragment-packed LDS)
__device__ __forceinline__ v8f gemm128(const float* A, int mt, int nt,
                                       const float* W, int lane) {
    v8f acc = {0.f, 0.f, 0.f, 0.f, 0.f, 0.f, 0.f, 0.f};
    const float* arow = A + (mt * 16 + (lane & 15)) * STRIDE + ((lane >> 4) << 1);
#pragma unroll 4
    for (int kt = 0; kt < 32; ++kt) {
        v2f a;
        a.x = arow[kt * 4];
        a.y = arow[kt * 4 + 1];
        v2f b = *(const v2f*)(W + ((kt * 8 + nt) << 6) + (lane << 1));
        acc = wmma4(a, b, acc);
    }
    return acc;
}

__global__ __launch_bounds__(NTHREADS, 1)
void schnet_fused(const int* __restrict__ zg, const float* __restrict__ pos,
                  const float* __restrict__ emb,
                  const float* __restrict__ fw1, const float* __restrict__ fb1,
                  const float* __restrict__ fw2, const float* __restrict__ fb2,
                  const float* __restrict__ l1w,
                  const float* __restrict__ l2w, const float* __restrict__ l2b,
                  const float* __restrict__ ow,  const float* __restrict__ ob,
                  const float* __restrict__ finw, const float* __restrict__ finb,
                  float* __restrict__ out)
{
    __shared__ float smem[SMEM_TOT];
    const int mol  = blockIdx.x;
    const int tid  = threadIdx.x;
    const int wave = tid >> 5;
    const int lane = tid & 31;
    const int mt   = wave >> 3;           // node-GEMM tile (16 waves = 2x8 tiles)
    const int nt   = wave & 7;
    const int lcol = lane & 15;
    const int lrow = (lane >> 4) << 3;    // 0 or 8 (C/D row base)
    const int kh   = (lane >> 4) << 1;    // 0 or 2 (A/B k base)

    float* const wb0 = &smem[WB0_OFF];
    float* const wb1 = &smem[WB1_OFF];

    // ---- init: embedding gather, positions ----
    for (int idx = tid; idx < NA * HID; idx += NTHREADS) {
        int a = idx >> 7, c = idx & 127;
        smem[SH_OFF + a * STRIDE + c] = emb[zg[mol * NA + a] * HID + c];
    }
    for (int idx = tid; idx < 8 * HID; idx += NTHREADS) {
        int a = NA + (idx >> 7), c = idx & 127;
        smem[SH_OFF + a * STRIDE + c] = 0.0f;    // pad rows of h
    }
    if (tid < NA * 3) smem[SP_OFF + tid] = pos[mol * NA * 3 + tid];
    __syncthreads();

    // ---- distances + cosine cutoff per slot (s = src*24 + dst) ----
    for (int s = tid; s < NS; s += NTHREADS) {
        int i = s / NA, j = s - i * NA;
        float dx = smem[SP_OFF + j * 3 + 0] - smem[SP_OFF + i * 3 + 0];
        float dy = smem[SP_OFF + j * 3 + 1] - smem[SP_OFF + i * 3 + 1];
        float dz = smem[SP_OFF + j * 3 + 2] - smem[SP_OFF + i * 3 + 2];
        float d  = sqrtf(dx * dx + dy * dy + dz * dz);
        smem[SD_OFF + s] = d;
        smem[SC_OFF + s] = (i == j) ? 0.0f
                                    : 0.5f * (__cosf(d * 0.3141592653f) + 1.0f);
    }
    // prologue: stage lin1(t=0) into buffer 0
    stage_w_full(wb0, l1w, 32, tid);
    __syncthreads();

    v8f zero8 = {0.f, 0.f, 0.f, 0.f, 0.f, 0.f, 0.f, 0.f};

    // ---- interaction blocks (double-buffered weights: 4 barriers / block) ----
    for (int t = 0; t < 6; ++t) {
        // == S1: xf = h @ lin1 (wb0) | stage w1,w2 -> wb1 | zero agg ==
        {
            v8f acc = gemm128(&smem[SH_OFF], mt, nt, wb0, lane);
            int colg = nt * 16 + lcol, rbase = mt * 16 + lrow;
#pragma unroll
            for (int r = 0; r < 8; ++r)
                smem[SXV_OFF + (rbase + r) * STRIDE + colg] = acc[r];
        }
        stage_w1(wb1, fw1 + t * 50 * HID, tid);
        stage_w_full(wb1 + 13 * 512, fw2 + t * HID * HID, 32, tid);
        for (int idx = tid; idx < 32 * STRIDE; idx += NTHREADS)
            smem[SAGG_OFF + idx] = 0.0f;
        __syncthreads();

        // == S2: edge filter net + message + scatter (wb1) | stage lin2 -> wb0 ==
        {
            const float* fb1t = fb1 + t * HID;
            const float* fb2t = fb2 + t * HID;
            // per-lane bias preload (registers), reused across slot tiles
            float bia0[4], bia1[4], bib[8];
#pragma unroll
            for (int c = 0; c < 4; ++c) {
                bia0[c] = fb1t[c * 32 + lcol];
                bia1[c] = fb1t[c * 32 + 16 + lcol];
            }
#pragma unroll
            for (int q = 0; q < 8; ++q) bib[q] = fb2t[q * 16 + lcol];

            // warm L2/WGP$ for next interaction's big weight matrices
            if (t < 5) {
                __builtin_prefetch(fw2 + (t + 1) * HID * HID + (tid << 5));
                __builtin_prefetch(l1w + (t + 1) * HID * HID + (tid << 5));
            }

            float* ub = &smem[SU_OFF + wave * 16 * USTR];
            const float* w1p = wb1;
            const float* w2p = wb1 + 13 * 512;
#pragma unroll 1
            for (int st = wave; st < NST; st += NWAVES) {
                int slot0 = st * 16;
                float dval = smem[SD_OFF + slot0 + lcol];   // A row = lane&15
                // rbf A-fragments, computed once, reused across 4 k-chunks
                v2f afrag[13];
#pragma unroll
                for (int kt = 0; kt < 13; ++kt) {
                    int g = kt * 4 + kh;
                    afrag[kt].x = rbff(dval, g);
                    afrag[kt].y = rbff(dval, g + 1);
                }
                v8f acc2[8];
#pragma unroll
                for (int q = 0; q < 8; ++q) acc2[q] = zero8;

#pragma unroll 1
                for (int c = 0; c < 4; ++c) {    // 2 n-tiles of GEMM1 per chunk
                    v8f u0 = zero8, u1 = zero8;
#pragma unroll
                    for (int kt = 0; kt < 13; ++kt) {   // K = 52 (padded 50)
                        v2f b0 = *(const v2f*)(w1p + ((kt * 8 + 2 * c) << 6) + (lane << 1));
                        v2f b1 = *(const v2f*)(w1p + ((kt * 8 + 2 * c + 1) << 6) + (lane << 1));
                        u0 = wmma4(afrag[kt], b0, u0);
                        u1 = wmma4(afrag[kt], b1, u1);
                    }
                    // bias + ssp, stage U chunk [16 x 32] (wave-private)
#pragma unroll
                    for (int r = 0; r < 8; ++r) {
                        ub[(lrow + r) * USTR + lcol]      = sspf(u0[r] + bia0[c]);
                        ub[(lrow + r) * USTR + 16 + lcol] = sspf(u1[r] + bia1[c]);
                    }
                    // GEMM2 partial accumulation over k = c*32 .. c*32+31
                    const float* urow = ub + lcol * USTR + kh;
#pragma unroll
                    for (int kt2 = 0; kt2 < 8; ++kt2) {
                        v2f a;
                        a.x = urow[kt2 * 4];
                        a.y = urow[kt2 * 4 + 1];
                        int ktg = c * 8 + kt2;
#pragma unroll
                        for (int q = 0; q < 8; ++q) {
                            v2f b = *(const v2f*)(w2p + ((ktg * 8 + q) << 6) + (lane << 1));
                            acc2[q] = wmma4(a, b, acc2[q]);
                        }
                    }
                }
                // per-row metadata (uniform across the 16 lanes of a half-wave)
                float cv[8];
                int soff[8], doff[8];
#pragma unroll
                for (int r = 0; r < 8; ++r) {
                    int s = slot0 + lrow + r;
                    cv[r] = smem[SC_OFF + s];
                    int si = s / NA;
                    soff[r] = si * STRIDE;
                    doff[r] = (s - si * NA) * STRIDE;
                }
                // W = (T2 + b2) * C ; msg = xf[src] * W ; agg[dst] += msg
#pragma unroll
                for (int q = 0; q < 8; ++q) {
                    int col = q * 16 + lcol;
#pragma unroll
                    for (int r = 0; r < 8; ++r) {
                        float Wv = (acc2[q][r] + bib[q]) * cv[r];
                        float m = smem[SXV_OFF + soff[r] + col] * Wv;
                        atomicAdd(&smem[SAGG_OFF + doff[r] + col], m);
                    }
                }
            }
        }
        stage_w_full(wb0, l2w + t * HID * HID, 32, tid);
        __syncthreads();

        // == S3: v = ssp(agg @ lin2 + b2) (wb0) | stage out_w -> wb1 ==
        {
            v8f acc = gemm128(&smem[SAGG_OFF], mt, nt, wb0, lane);
            int colg = nt * 16 + lcol, rbase = mt * 16 + lrow;
            float bv = l2b[t * HID + colg];
#pragma unroll
            for (int r = 0; r < 8; ++r)
                smem[SXV_OFF + (rbase + r) * STRIDE + colg] = sspf(acc[r] + bv);
        }
        stage_w_full(wb1, ow + t * HID * HID, 32, tid);
        __syncthreads();

        // == S4: h += v @ out_w + ob (wb1) | stage lin1(t+1) -> wb0 ==
        {
            v8f acc = gemm128(&smem[SXV_OFF], mt, nt, wb1, lane);
            int colg = nt * 16 + lcol, rbase = mt * 16 + lrow;
            float bv = ob[t * HID + colg];
#pragma unroll
            for (int r = 0; r < 8; ++r)
                smem[SH_OFF + (rbase + r) * STRIDE + colg] += acc[r] + bv;
        }
        if (t < 5) stage_w_full(wb0, l1w + (t + 1) * HID * HID, 32, tid);
        __syncthreads();
    }

    // ---- readout: pooled = sum_atoms h ; out = pooled . final_w + final_b ----
    if (tid < HID) {
        float s = 0.0f;
#pragma unroll
        for (int a = 0; a < NA; ++a) s += smem[SH_OFF + a * STRIDE + tid];
        smem[SD_OFF + tid] = s * finw[tid];
    }
    __syncthreads();
    if (tid == 0) {
        float s = finb[0];
        for (int c = 0; c < HID; ++c) s += smem[SD_OFF + c];
        out[mol] = s;
    }
}

extern "C" void kernel_launch(void* const* d_in, const int* in_sizes, int n_in,
                              void* d_out, int out_size, void* d_ws, size_t ws_size,
                              hipStream_t stream) {
    const int*   z    = (const int*)d_in[0];
    const float* pos  = (const float*)d_in[1];
    // d_in[2] = batch (implicit), d_in[3] = edge_index (implicit) -- unused
    const float* emb  = (const float*)d_in[4];
    const float* fw1  = (const float*)d_in[5];
    const float* fb1  = (const float*)d_in[6];
    const float* fw2  = (const float*)d_in[7];
    const float* fb2  = (const float*)d_in[8];
    const float* l1w  = (const float*)d_in[9];
    const float* l2w  = (const float*)d_in[10];
    const float* l2b  = (const float*)d_in[11];
    const float* ow   = (const float*)d_in[12];
    const float* ob   = (const float*)d_in[13];
    const float* finw = (const float*)d_in[14];
    const float* finb = (const float*)d_in[15];

    int nmol = in_sizes[0] / NA;   // 1024
    schnet_fused<<<nmol, NTHREADS, 0, stream>>>(z, pos, emb, fw1, fb1, fw2, fb2,
                                                l1w, l2w, l2b, ow, ob, finw, finb,
                                                (float*)d_out);
}